// GCN_39058432590069
// MI455X (gfx1250) — compile-verified
//
#include <hip/hip_runtime.h>

typedef float v2f __attribute__((ext_vector_type(2)));
typedef float v8f __attribute__((ext_vector_type(8)));

static constexpr int KDIM = 128;     // inner dim for both layers
static constexpr int ROWS = 64;      // node rows per GEMM block
static constexpr int KC   = 64;      // K chunk staged in LDS
static constexpr int LSTR = KC + 4;  // padded LDS stride (68 dwords): bank-conflict-free b64

// ---------------------------------------------------------------------------
// support = X @ W using V_WMMA_F32_16X16X4_F32 (exact fp32 matrix op).
// 256 threads = 8 waves. 64 rows x NDIM cols per block, K staged in 2 chunks.
// ---------------------------------------------------------------------------
template<int NDIM, bool RELU_A>
__global__ __launch_bounds__(256)
void gemm_xw_wmma(const float* __restrict__ X, const float* __restrict__ W,
                  float* __restrict__ Out, int n_rows)
{
    constexpr int CT  = NDIM / 16;   // 16-col tiles (8 or 4)
    constexpr int WPC = 8 / CT;      // waves per col tile (1 or 2)
    constexpr int MT  = 4 / WPC;     // 16-row M tiles per wave (4 or 2)

    __shared__ float lds_a[ROWS * LSTR];   // 64x68 f32 = 17,408 B
    __shared__ float lds_w[NDIM * LSTR];   // <= 128x68 f32 = 34,816 B

    const int row0 = blockIdx.x * ROWS;
    const int tid  = threadIdx.x;
    const int lane = tid & 31;
    const int wave = tid >> 5;
    const int ln   = lane & 15;
    const int koff = (lane >> 4) << 1;      // lanes 0-15 -> K+0/1, lanes 16-31 -> K+2/3
    const int col_tile = wave % CT;
    const int m_begin  = (wave / CT) * MT;

    v8f acc[MT];
#pragma unroll
    for (int t = 0; t < MT; ++t)
        acc[t] = (v8f){0.f,0.f,0.f,0.f,0.f,0.f,0.f,0.f};

    for (int kc = 0; kc < KDIM; kc += KC) {
        __syncthreads();
        // ---- stage A chunk: lds_a[r*LSTR + k] = X[row0+r][kc+k] (ReLU fused) ----
        for (int i = tid; i < ROWS * (KC/4); i += 256) {
            int r  = i / (KC/4);
            int c4 = (i % (KC/4)) * 4;
            float4 v = make_float4(0.f, 0.f, 0.f, 0.f);
            if (row0 + r < n_rows)
                v = *reinterpret_cast<const float4*>(X + (size_t)(row0 + r) * KDIM + kc + c4);
            if (RELU_A) {
                v.x = fmaxf(v.x, 0.f); v.y = fmaxf(v.y, 0.f);
                v.z = fmaxf(v.z, 0.f); v.w = fmaxf(v.w, 0.f);
            }
            *reinterpret_cast<float4*>(&lds_a[r * LSTR + c4]) = v;
        }
        // ---- stage W^T chunk: lds_w[n*LSTR + k] = W[kc+k][n] ----
        for (int i = tid; i < KC * (NDIM/4); i += 256) {
            int k  = i / (NDIM/4);
            int n4 = (i % (NDIM/4)) * 4;
            float4 v = *reinterpret_cast<const float4*>(W + (size_t)(kc + k) * NDIM + n4);
            lds_w[(n4 + 0) * LSTR + k] = v.x;
            lds_w[(n4 + 1) * LSTR + k] = v.y;
            lds_w[(n4 + 2) * LSTR + k] = v.z;
            lds_w[(n4 + 3) * LSTR + k] = v.w;
        }
        __syncthreads();

        const float* bp = &lds_w[(col_tile * 16 + ln) * LSTR + koff];
        const float* ap = &lds_a[(m_begin * 16 + ln) * LSTR + koff];
#pragma unroll 4
        for (int k0 = 0; k0 < KC; k0 += 4) {
            v2f b = *reinterpret_cast<const v2f*>(bp + k0);
#pragma unroll
            for (int t = 0; t < MT; ++t) {
                v2f a = *reinterpret_cast<const v2f*>(ap + t * 16 * LSTR + k0);
                acc[t] = __builtin_amdgcn_wmma_f32_16x16x4_f32(
                    false, a, false, b, (short)0, acc[t], false, false);
            }
        }
    }

    // D layout: VGPR r holds row (r + (lane>>4)*8), col = lane&15 within tile
    const int col   = col_tile * 16 + ln;
    const int rbase = (lane >> 4) * 8;
#pragma unroll
    for (int t = 0; t < MT; ++t) {
        int m0 = row0 + (m_begin + t) * 16 + rbase;
#pragma unroll
        for (int r = 0; r < 8; ++r) {
            int row = m0 + r;
            if (row < n_rows)
                Out[(size_t)row * NDIM + col] = acc[t][r];
        }
    }
}

// ---------------------------------------------------------------------------
// Degree / normalization
// ---------------------------------------------------------------------------
__global__ void deg_init_kernel(float* __restrict__ deg, int n) {
    int i = blockIdx.x * blockDim.x + threadIdx.x;
    if (i < n) deg[i] = 1.0f;   // self loop
}

__global__ void deg_count_kernel(const long long* __restrict__ dst,
                                 float* __restrict__ deg, int e) {
    int i = blockIdx.x * blockDim.x + threadIdx.x;
    if (i < e) unsafeAtomicAdd(&deg[(int)dst[i]], 1.0f);
}

__global__ void deg_rsqrt_kernel(float* __restrict__ deg, int n) {
    int i = blockIdx.x * blockDim.x + threadIdx.x;
    if (i < n) deg[i] = rsqrtf(deg[i]);
}

// ---------------------------------------------------------------------------
// agg init with self-loop term: agg[i] = support[i] * dinv[i]^2
// ---------------------------------------------------------------------------
template<int C>
__global__ void spmm_self_kernel(const float* __restrict__ support,
                                 const float* __restrict__ dinv,
                                 float* __restrict__ out, int n) {
    constexpr int L = C / 4;
    unsigned int i = blockIdx.x * blockDim.x + threadIdx.x;
    unsigned int total = (unsigned int)n * L;
    if (i >= total) return;
    int row = i / L;
    int c4  = (i % L) * 4;
    float s = dinv[row]; s *= s;
    float4 v = *reinterpret_cast<const float4*>(support + (size_t)row * C + c4);
    v.x *= s; v.y *= s; v.z *= s; v.w *= s;
    *reinterpret_cast<float4*>(out + (size_t)row * C + c4) = v;
}

// ---------------------------------------------------------------------------
// Edge scatter: agg[dst] += support[src] * dinv[src]*dinv[dst]
// One lane handles 4 columns -> coalesced float4 gather + 4 f32 atomics (L2).
// ---------------------------------------------------------------------------
template<int C>
__global__ void spmm_edge_kernel(const long long* __restrict__ src,
                                 const long long* __restrict__ dst,
                                 const float* __restrict__ support,
                                 const float* __restrict__ dinv,
                                 float* __restrict__ out, int e) {
    constexpr int L = C / 4;
    unsigned int gid = blockIdx.x * blockDim.x + threadIdx.x;
    unsigned int total = (unsigned int)e * L;
    if (gid >= total) return;
    int eidx = gid / L;
    int c4   = (gid % L) * 4;
    int s = (int)src[eidx];
    int d = (int)dst[eidx];
    float val = dinv[s] * dinv[d];
    float4 v = *reinterpret_cast<const float4*>(support + (size_t)s * C + c4);
    float* o = out + (size_t)d * C + c4;
    unsafeAtomicAdd(o + 0, v.x * val);
    unsafeAtomicAdd(o + 1, v.y * val);
    unsafeAtomicAdd(o + 2, v.z * val);
    unsafeAtomicAdd(o + 3, v.w * val);
}

// ---------------------------------------------------------------------------
extern "C" void kernel_launch(void* const* d_in, const int* in_sizes, int n_in,
                              void* d_out, int out_size, void* d_ws, size_t ws_size,
                              hipStream_t stream)
{
    const float*     x    = (const float*)d_in[0];
    const long long* edge = (const long long*)d_in[1];   // [2, E] int64
    const float*     W1   = (const float*)d_in[2];       // [128,128]
    const float*     W2   = (const float*)d_in[3];       // [128,64]

    const int n = in_sizes[0] / KDIM;     // 100000
    const int e = in_sizes[1] / 2;        // 1600000
    const long long* srcI = edge;
    const long long* dstI = edge + e;

    // workspace layout (256B aligned)
    char* ws = (char*)d_ws;
    size_t o0 = 0;                                      // dinv: n f32
    size_t o1 = (o0 + (size_t)n * 4 + 255) & ~(size_t)255;  // support1: n*128 f32
    size_t o2 = o1 + (size_t)n * KDIM * 4;                  // h (agg1):  n*128 f32
    size_t o3 = o2 + (size_t)n * KDIM * 4;                  // support2:  n*64 f32
    float* dinv = (float*)(ws + o0);
    float* sup1 = (float*)(ws + o1);
    float* h    = (float*)(ws + o2);
    float* sup2 = (float*)(ws + o3);
    float* out  = (float*)d_out;

    const int T = 256;
    // normalization d^{-1/2} (shared by both layers)
    deg_init_kernel<<<(n + T - 1) / T, T, 0, stream>>>(dinv, n);
    deg_count_kernel<<<(e + T - 1) / T, T, 0, stream>>>(dstI, dinv, e);
    deg_rsqrt_kernel<<<(n + T - 1) / T, T, 0, stream>>>(dinv, n);

    // layer 1: support1 = x @ W1 ; h = A_hat * support1
    gemm_xw_wmma<128, false><<<(n + ROWS - 1) / ROWS, T, 0, stream>>>(x, W1, sup1, n);
    {
        unsigned int tot_s = (unsigned int)n * 32;
        unsigned int tot_e = (unsigned int)e * 32;
        spmm_self_kernel<128><<<(tot_s + T - 1) / T, T, 0, stream>>>(sup1, dinv, h, n);
        spmm_edge_kernel<128><<<(tot_e + T - 1) / T, T, 0, stream>>>(srcI, dstI, sup1, dinv, h, e);
    }

    // layer 2: support2 = relu(h) @ W2 (ReLU fused into A staging); out = A_hat * support2
    gemm_xw_wmma<64, true><<<(n + ROWS - 1) / ROWS, T, 0, stream>>>(h, W2, sup2, n);
    {
        unsigned int tot_s = (unsigned int)n * 16;
        unsigned int tot_e = (unsigned int)e * 16;
        spmm_self_kernel<64><<<(tot_s + T - 1) / T, T, 0, stream>>>(sup2, dinv, out, n);
        spmm_edge_kernel<64><<<(tot_e + T - 1) / T, T, 0, stream>>>(srcI, dstI, sup2, dinv, out, e);
    }
}